// DirectedGAT_6811818131851
// MI455X (gfx1250) — compile-verified
//
#include <hip/hip_runtime.h>
#include <hip/hip_bf16.h>

typedef __attribute__((ext_vector_type(2)))  float        v2f;
typedef __attribute__((ext_vector_type(8)))  float        v8f;
typedef __attribute__((ext_vector_type(4)))  unsigned int u32x4;
typedef __attribute__((ext_vector_type(8)))  int          i32x8;
typedef __attribute__((ext_vector_type(4)))  int          i32x4;

#define NEG_SLOPE 0.2f
#define MAX_K 512
// LDS floats needed: 16*MAX_K + pad(4 dwords per 256 dwords)
#define A_LDS_FLOATS (16 * MAX_K + (16 * MAX_K / 256) * 4)   // 8320

#if __has_builtin(__builtin_amdgcn_tensor_load_to_lds) && __has_builtin(__builtin_amdgcn_s_wait_tensorcnt)
#define HAVE_TDM 1
#else
#define HAVE_TDM 0
#endif

// Padded LDS index: TDM inserts 4 dwords of pad after every 256 dwords stored.
__device__ __forceinline__ int lds_pad_idx(int flat) {
  return flat + ((flat >> 8) << 2);
}

// ---------------------------------------------------------------------------
// GEMM: C[M x Nout] = A[M x K] * W[Nout x K]^T   (h = x @ W.T)
// Block = 4 waves (128 thr). Block owns one 16-row A panel, staged once into
// LDS by the Tensor Data Mover; each wave computes a 16x16 tile with fp32
// WMMA 16x16x4, A from LDS (ds_load_b64, bank-conflict-free via D# padding),
// B streamed from global (L2-resident).  grid = (M/16, Nout/64)
// ---------------------------------------------------------------------------
__global__ void gemm_xWt_wmma(const float* __restrict__ A,
                              const float* __restrict__ W,
                              float* __restrict__ C,
                              int M, int K, int Nout) {
  __shared__ float aT[A_LDS_FLOATS];

  const int lane = threadIdx.x & 31;
  const int wave = threadIdx.x >> 5;
  const int tm = blockIdx.x;                 // M tile
  const int tn = blockIdx.y * 4 + wave;      // N tile (uniform per wave)

#if HAVE_TDM
  if (wave == 0) {
    // ---- Tensor DMA descriptor (D#) per CDNA5 ISA ----
    unsigned lds_base = (unsigned)(unsigned long long)(&aT[0]);  // addr[31:0] = LDS byte offset
    unsigned long long ga =
        (unsigned long long)(const void*)(A + (size_t)tm * 16 * K);
    u32x4 g0;
    g0[0] = 1u;                                   // count=1 (valid), flags 0
    g0[1] = lds_base;                             // lds_addr
    g0[2] = (unsigned)ga;                         // global_addr[31:0]
    g0[3] = (unsigned)((ga >> 32) & 0x01FFFFFFu)  // global_addr[56:32]
            | 0x80000000u;                        // type=2 ("image")
    i32x8 g1;
    // data_size=2 (4B) | pad_enable | pad_interval=7 (256 dw) | pad_amount=3 (4 dw)
    g1[0] = (int)(0x00020000u | (1u << 20) | (7u << 22) | (3u << 25));
    g1[1] = (int)((unsigned)(K & 0xFFFF) << 16);  // tensor_dim0[15:0] (atomic_barrier=0)
    g1[2] = (int)(16u << 16);                     // tensor_dim0[31:16]=0 | tensor_dim1[15:0]=16
    g1[3] = (int)((unsigned)K << 16);             // tensor_dim1[31:16]=0 | tile_dim0=K
    g1[4] = 16;                                   // tile_dim1=16 | tile_dim2=0
    g1[5] = K;                                    // tensor_dim0_stride[31:0]
    g1[6] = 0;                                    // stride0[47:32]=0 | stride1[15:0]=0
    g1[7] = 0;
    i32x4 gz = {0, 0, 0, 0};                      // 2D tile: groups 2/3 unused
#if __clang_major__ >= 23
    i32x8 gz8 = {0, 0, 0, 0, 0, 0, 0, 0};
    __builtin_amdgcn_tensor_load_to_lds(g0, g1, gz, gz, gz8, 0);
#else
    __builtin_amdgcn_tensor_load_to_lds(g0, g1, gz, gz, 0);
#endif
    __builtin_amdgcn_s_wait_tensorcnt(0);
  }
  __syncthreads();
#else
  // Cooperative LDS staging fallback (same padded layout).
  for (int i = threadIdx.x; i < 16 * K; i += blockDim.x) {
    int r = i / K, k = i - r * K;
    aT[lds_pad_idx(r * K + k)] = A[(size_t)(tm * 16 + r) * K + k];
  }
  __syncthreads();
#endif

  const int l15  = lane & 15;
  const int hi   = lane >> 4;        // 0: lanes 0-15, 1: lanes 16-31
  const int koff = hi * 2;           // A/B fragment K pair: lanes 0-15 -> k0+0,1; 16-31 -> k0+2,3
  const int rowflat = l15 * K;
  const float* __restrict__ wrow = W + (size_t)(tn * 16 + l15) * K;

  v8f acc = {};
#pragma unroll 4
  for (int k0 = 0; k0 < K; k0 += 4) {
    int idx = lds_pad_idx(rowflat + k0 + koff);
    v2f a = *(const v2f*)(&aT[idx]);
    v2f b = *(const v2f*)(wrow + k0 + koff);
    acc = __builtin_amdgcn_wmma_f32_16x16x4_f32(
        /*neg_a=*/false, a, /*neg_b=*/false, b,
        /*c_mod=*/(short)0, acc, /*reuse_a=*/false, /*reuse_b=*/false);
  }

  // D layout (16x16 f32): VGPR i -> row tm*16 + i + 8*hi, col = tn*16 + l15
  const int col   = tn * 16 + l15;
  const int rbase = tm * 16 + hi * 8;
#pragma unroll
  for (int i = 0; i < 8; ++i)
    C[(size_t)(rbase + i) * Nout + col] = acc[i];
}

// ---------------------------------------------------------------------------
// Attention logits: als[n,h] = <h[n,h,:], a_src[h,:]> ; ald likewise.
// ---------------------------------------------------------------------------
__global__ void attn_logits(const float* __restrict__ ht,
                            const float* __restrict__ a_src,
                            const float* __restrict__ a_dst,
                            float* __restrict__ als, float* __restrict__ ald,
                            int n_nodes, int heads, int C) {
  int idx = blockIdx.x * blockDim.x + threadIdx.x;
  if (idx >= n_nodes * heads) return;
  int n = idx / heads, h = idx - n * heads;
  const float* hp = ht + (size_t)n * heads * C + (size_t)h * C;
  const float* as = a_src + h * C;
  const float* ad = a_dst + h * C;
  float s = 0.f, d = 0.f;
  for (int c = 0; c < C; ++c) { float v = hp[c]; s += v * as[c]; d += v * ad[c]; }
  als[idx] = s; ald[idx] = d;
}

__global__ void fill_f32(float* __restrict__ p, float v, size_t n) {
  size_t i = (size_t)blockIdx.x * blockDim.x + threadIdx.x;
  size_t stride = (size_t)gridDim.x * blockDim.x;
  for (; i < n; i += stride) p[i] = v;
}

__device__ __forceinline__ void atomicMaxF32(float* addr, float val) {
  if (val >= 0.0f) atomicMax((int*)addr, __float_as_int(val));
  else             atomicMin((unsigned int*)addr, __float_as_uint(val));
}

__device__ __forceinline__ void edge_src_dst(const int* __restrict__ ei, int E,
                                             int e, int rev, int& src, int& dst) {
  if (e < E) { src = ei[e]; dst = ei[E + e]; }
  else       { src = dst = e - E; }                 // self loop
  if (rev) { int t = src; src = dst; dst = t; }     // reversed (out) direction
}

__device__ __forceinline__ float leaky(float x) {
  return x > 0.0f ? x : NEG_SLOPE * x;
}

// ---------------------------------------------------------------------------
// Edge pass 1: per-destination max of leaky_relu(als[src]+ald[dst])
// ---------------------------------------------------------------------------
__global__ void edge_segmax(const int* __restrict__ ei, int E, int Nn, int rev,
                            const float* __restrict__ als,
                            const float* __restrict__ ald,
                            float* __restrict__ m, int heads) {
  size_t idx = (size_t)blockIdx.x * blockDim.x + threadIdx.x;
  size_t total = (size_t)(E + Nn) * heads;
  if (idx >= total) return;
  int e = (int)(idx / heads), h = (int)(idx - (size_t)e * heads);
  int src, dst; edge_src_dst(ei, E, e, rev, src, dst);
  float v = leaky(als[src * heads + h] + ald[dst * heads + h]);
  atomicMaxF32(&m[dst * heads + h], v);
}

// ---------------------------------------------------------------------------
// Edge pass 2: denom[dst] += exp(e - m[dst])
// ---------------------------------------------------------------------------
__global__ void edge_segsum(const int* __restrict__ ei, int E, int Nn, int rev,
                            const float* __restrict__ als,
                            const float* __restrict__ ald,
                            const float* __restrict__ m,
                            float* __restrict__ den, int heads) {
  size_t idx = (size_t)blockIdx.x * blockDim.x + threadIdx.x;
  size_t total = (size_t)(E + Nn) * heads;
  if (idx >= total) return;
  int e = (int)(idx / heads), h = (int)(idx - (size_t)e * heads);
  int src, dst; edge_src_dst(ei, E, e, rev, src, dst);
  int di = dst * heads + h;
  float v = leaky(als[src * heads + h] + ald[di]);
  atomicAdd(&den[di], __expf(v - m[di]));
}

// ---------------------------------------------------------------------------
// Edge pass 3: out[dst, obase + c] += alpha * h[src, c]
// 64 lanes x float4 per edge row (global_load_b128); 4 edges per 256-thr block.
// ---------------------------------------------------------------------------
__global__ void edge_aggregate(const int* __restrict__ ei, int E, int Nn, int rev,
                               const float* __restrict__ als,
                               const float* __restrict__ ald,
                               const float* __restrict__ m,
                               const float* __restrict__ den,
                               const float* __restrict__ ht,
                               float* __restrict__ out,
                               int heads, int C, int ostride, int obase) {
  const int HC  = heads * C;                 // 256
  const int sub = threadIdx.x >> 6;          // edge slot within block (0..3)
  const int t   = threadIdx.x & 63;
  const int c4  = t * 4;                     // channel group
  const int h   = c4 / C;
  const int Etot = E + Nn;
  for (int e = blockIdx.x * 4 + sub; e < Etot; e += gridDim.x * 4) {
    int src, dst; edge_src_dst(ei, E, e, rev, src, dst);
    int di = dst * heads + h;
    float v = leaky(als[src * heads + h] + ald[di]);
    float alpha = __expf(v - m[di]) / fmaxf(den[di], 1e-16f);
    const float4 hv = *(const float4*)(ht + (size_t)src * HC + c4);
    float* op = out + (size_t)dst * ostride + obase + c4;
    atomicAdd(op + 0, alpha * hv.x);
    atomicAdd(op + 1, alpha * hv.y);
    atomicAdd(op + 2, alpha * hv.z);
    atomicAdd(op + 3, alpha * hv.w);
  }
}

// ---------------------------------------------------------------------------
// buf[n, c] = act(buf[n, c] + (c < half ? b0[c] : b1[c-half]))
// ---------------------------------------------------------------------------
__global__ void bias_act(float* __restrict__ buf,
                         const float* __restrict__ b0,
                         const float* __restrict__ b1,
                         int Nn, int half, int do_relu) {
  size_t idx = (size_t)blockIdx.x * blockDim.x + threadIdx.x;
  size_t total = (size_t)Nn * 2 * half;
  if (idx >= total) return;
  int col = (int)(idx % (size_t)(2 * half));
  float bv = (col < half) ? b0[col] : b1[col - half];
  float v = buf[idx] + bv;
  if (do_relu) v = fmaxf(v, 0.0f);
  buf[idx] = v;
}

// ---------------------------------------------------------------------------
// Host orchestration
// ---------------------------------------------------------------------------
extern "C" void kernel_launch(void* const* d_in, const int* in_sizes, int n_in,
                              void* d_out, int out_size, void* d_ws, size_t ws_size,
                              hipStream_t stream) {
  const float* x      = (const float*)d_in[0];
  const int*   ei     = (const int*)d_in[1];
  const float* W_in0  = (const float*)d_in[2];
  const float* as_in0 = (const float*)d_in[3];
  const float* ad_in0 = (const float*)d_in[4];
  const float* b_in0  = (const float*)d_in[5];
  const float* W_out0 = (const float*)d_in[6];
  const float* as_out0= (const float*)d_in[7];
  const float* ad_out0= (const float*)d_in[8];
  const float* b_out0 = (const float*)d_in[9];
  const float* W_in1  = (const float*)d_in[10];
  const float* as_in1 = (const float*)d_in[11];
  const float* ad_in1 = (const float*)d_in[12];
  const float* b_in1  = (const float*)d_in[13];
  const float* W_out1 = (const float*)d_in[14];
  const float* as_out1= (const float*)d_in[15];
  const float* ad_out1= (const float*)d_in[16];
  const float* b_out1 = (const float*)d_in[17];
  float* out = (float*)d_out;

  const int IN_CH = 256, HID = 256;
  const int Nn = in_sizes[0] / IN_CH;      // 20000
  const int E  = in_sizes[1] / 2;          // 640000
  const int Etot = E + Nn;

  // Workspace carve-up (floats)
  float* ws = (float*)d_ws;
  size_t off = 0;
  float* ht_a  = ws + off; off += (size_t)Nn * HID;     // conv "in"  transform
  float* ht_b  = ws + off; off += (size_t)Nn * HID;     // conv "out" transform
  float* hcat  = ws + off; off += (size_t)Nn * 2 * HID; // layer-0 activation
  float* als_a = ws + off; off += (size_t)Nn * 4;
  float* ald_a = ws + off; off += (size_t)Nn * 4;
  float* als_b = ws + off; off += (size_t)Nn * 4;
  float* ald_b = ws + off; off += (size_t)Nn * 4;
  float* m_a   = ws + off; off += (size_t)Nn * 4;
  float* m_b   = ws + off; off += (size_t)Nn * 4;
  float* den_a = ws + off; off += (size_t)Nn * 4;
  float* den_b = ws + off; off += (size_t)Nn * 4;
  (void)ws_size; (void)n_in; (void)out_size;

  auto cdiv = [](size_t a, size_t b) { return (unsigned)((a + b - 1) / b); };
  const float NEG_INF = -1e30f;

  // =============================== Layer 0 (heads=4, C=64) =================
  {
    const int H = 4, C = 64, K = IN_CH, Nout = HID;
    dim3 gg(Nn / 16, Nout / 64);
    gemm_xWt_wmma<<<gg, 128, 0, stream>>>(x, W_in0,  ht_a, Nn, K, Nout);
    gemm_xWt_wmma<<<gg, 128, 0, stream>>>(x, W_out0, ht_b, Nn, K, Nout);

    attn_logits<<<cdiv((size_t)Nn * H, 256), 256, 0, stream>>>(ht_a, as_in0,  ad_in0,  als_a, ald_a, Nn, H, C);
    attn_logits<<<cdiv((size_t)Nn * H, 256), 256, 0, stream>>>(ht_b, as_out0, ad_out0, als_b, ald_b, Nn, H, C);

    fill_f32<<<512, 256, 0, stream>>>(m_a,   NEG_INF, (size_t)Nn * H);
    fill_f32<<<512, 256, 0, stream>>>(m_b,   NEG_INF, (size_t)Nn * H);
    fill_f32<<<512, 256, 0, stream>>>(den_a, 0.0f,    (size_t)Nn * H);
    fill_f32<<<512, 256, 0, stream>>>(den_b, 0.0f,    (size_t)Nn * H);
    fill_f32<<<2048, 256, 0, stream>>>(hcat, 0.0f,    (size_t)Nn * 2 * HID);

    unsigned ge = cdiv((size_t)Etot * H, 256);
    edge_segmax<<<ge, 256, 0, stream>>>(ei, E, Nn, 0, als_a, ald_a, m_a, H);
    edge_segmax<<<ge, 256, 0, stream>>>(ei, E, Nn, 1, als_b, ald_b, m_b, H);
    edge_segsum<<<ge, 256, 0, stream>>>(ei, E, Nn, 0, als_a, ald_a, m_a, den_a, H);
    edge_segsum<<<ge, 256, 0, stream>>>(ei, E, Nn, 1, als_b, ald_b, m_b, den_b, H);

    unsigned ga = cdiv((size_t)Etot, 4);
    edge_aggregate<<<ga, 256, 0, stream>>>(ei, E, Nn, 0, als_a, ald_a, m_a, den_a,
                                           ht_a, hcat, H, C, 2 * HID, 0);
    edge_aggregate<<<ga, 256, 0, stream>>>(ei, E, Nn, 1, als_b, ald_b, m_b, den_b,
                                           ht_b, hcat, H, C, 2 * HID, HID);

    bias_act<<<cdiv((size_t)Nn * 2 * HID, 256), 256, 0, stream>>>(hcat, b_in0, b_out0, Nn, HID, 1);
  }

  // =============================== Layer 1 (heads=1, C=256) ================
  {
    const int H = 1, C = 256, K = 2 * HID, Nout = HID;
    dim3 gg(Nn / 16, Nout / 64);
    gemm_xWt_wmma<<<gg, 128, 0, stream>>>(hcat, W_in1,  ht_a, Nn, K, Nout);
    gemm_xWt_wmma<<<gg, 128, 0, stream>>>(hcat, W_out1, ht_b, Nn, K, Nout);

    attn_logits<<<cdiv((size_t)Nn * H, 256), 256, 0, stream>>>(ht_a, as_in1,  ad_in1,  als_a, ald_a, Nn, H, C);
    attn_logits<<<cdiv((size_t)Nn * H, 256), 256, 0, stream>>>(ht_b, as_out1, ad_out1, als_b, ald_b, Nn, H, C);

    fill_f32<<<512, 256, 0, stream>>>(m_a,   NEG_INF, (size_t)Nn * H);
    fill_f32<<<512, 256, 0, stream>>>(m_b,   NEG_INF, (size_t)Nn * H);
    fill_f32<<<512, 256, 0, stream>>>(den_a, 0.0f,    (size_t)Nn * H);
    fill_f32<<<512, 256, 0, stream>>>(den_b, 0.0f,    (size_t)Nn * H);
    fill_f32<<<2048, 256, 0, stream>>>(out,  0.0f,    (size_t)Nn * 2 * HID);

    unsigned ge = cdiv((size_t)Etot * H, 256);
    edge_segmax<<<ge, 256, 0, stream>>>(ei, E, Nn, 0, als_a, ald_a, m_a, H);
    edge_segmax<<<ge, 256, 0, stream>>>(ei, E, Nn, 1, als_b, ald_b, m_b, H);
    edge_segsum<<<ge, 256, 0, stream>>>(ei, E, Nn, 0, als_a, ald_a, m_a, den_a, H);
    edge_segsum<<<ge, 256, 0, stream>>>(ei, E, Nn, 1, als_b, ald_b, m_b, den_b, H);

    unsigned ga = cdiv((size_t)Etot, 4);
    edge_aggregate<<<ga, 256, 0, stream>>>(ei, E, Nn, 0, als_a, ald_a, m_a, den_a,
                                           ht_a, out, H, C, 2 * HID, 0);
    edge_aggregate<<<ga, 256, 0, stream>>>(ei, E, Nn, 1, als_b, ald_b, m_b, den_b,
                                           ht_b, out, H, C, 2 * HID, HID);

    bias_act<<<cdiv((size_t)Nn * 2 * HID, 256), 256, 0, stream>>>(out, b_in1, b_out1, Nn, HID, 0);
  }
}